// GNNClassifier_87737591923116
// MI455X (gfx1250) — compile-verified
//
#include <hip/hip_runtime.h>
#include <math.h>

#define NODES 100000
#define EDGES 1600000
#define ETOT  (EDGES + NODES)
#define DIN   128
#define HIDC  64
#define H1    4
#define F1    (H1 * HIDC)   // 256
#define NEG_SLOPE 0.2f
#define BN_EPS 1e-5f
#define KMAX  256           // max K across all GEMMs (LDS B-panel sizing)

typedef __attribute__((ext_vector_type(2))) float v2f;
typedef __attribute__((ext_vector_type(8))) float v8f;

static inline int cdiv(long long a, long long b) { return (int)((a + b - 1) / b); }

__device__ inline void atomicMaxFloat(float* addr, float val) {
    // valid when initialized to -inf: non-negative floats order as signed ints,
    // negative floats order reversed as unsigned ints
    if (val >= 0.0f) atomicMax((int*)addr, __float_as_int(val));
    else             atomicMin((unsigned int*)addr, __float_as_uint(val));
}

// ---------------- WMMA f32 GEMM: C[M,N] = A[M,K] @ B[K,N] (+bias)(+ELU) ------
// 8 waves per block; each wave owns one 16x16 output tile (same N-tile column).
// The K x 16 B panel is staged once per workgroup into LDS with CDNA5 async
// global->LDS copies (ASYNCcnt), then each wave runs v_wmma_f32_16x16x4_f32
// over K in steps of 4, reading B from LDS.
template <int ACT>
__global__ __launch_bounds__(256) void wmma_gemm_f32(const float* __restrict__ A,
                                                     const float* __restrict__ B,
                                                     const float* __restrict__ bias,
                                                     float* __restrict__ C,
                                                     int K, int Ncols, int Mtiles) {
    __shared__ float Bs[KMAX * 16];
    const int tid = threadIdx.x;
    const int tile_n = blockIdx.y;

    // ---- async-stage B panel (K x 16) into LDS ----
    // K*16 is always a multiple of 256 here, so every iteration has full EXEC.
    for (int idx = tid; idx < K * 16; idx += 256) {
        const int k = idx >> 4;
        const int c = idx & 15;
        const float* gp = B + (size_t)k * Ncols + (tile_n * 16 + c);
        const unsigned lds_off = (unsigned)(size_t)(&Bs[idx]);  // LDS offset: low 32 bits of the pointer
        asm volatile("global_load_async_to_lds_b32 %0, %1, off"
                     :: "v"(lds_off), "v"((unsigned long long)gp)
                     : "memory");
    }
    asm volatile("s_wait_asynccnt 0" ::: "memory");
    __syncthreads();

    // ---- per-wave 16x16 tile via v_wmma_f32_16x16x4_f32 ----
    const int wave = tid >> 5;
    const int lane = tid & 31;
    const int half = lane >> 4;        // 0: K={k,k+1}, 1: K={k+2,k+3}
    const int l    = lane & 15;
    const int tile_m = blockIdx.x * 8 + wave;
    if (tile_m < Mtiles) {             // wave-uniform guard: EXEC all-ones inside
        const int row = tile_m * 16 + l;
        const int col = tile_n * 16 + l;
        const float* arow = A + (size_t)row * K;
        v8f acc = {};
        for (int k0 = 0; k0 < K; k0 += 4) {
            const int ka = k0 + half * 2;
            v2f a, b;
            a.x = arow[ka];
            a.y = arow[ka + 1];
            b.x = Bs[ka * 16 + l];
            b.y = Bs[(ka + 1) * 16 + l];
            acc = __builtin_amdgcn_wmma_f32_16x16x4_f32(false, a, false, b,
                                                        (short)0, acc, false, false);
        }
        const float bv = bias ? bias[col] : 0.0f;
#pragma unroll
        for (int r = 0; r < 8; ++r) {
            const int orow = tile_m * 16 + r + half * 8;
            float v = acc[r] + bv;
            if (ACT == 1) v = (v > 0.0f) ? v : (expf(v) - 1.0f);
            C[(size_t)orow * Ncols + col] = v;
        }
    }
}

// ---------------- attention scores: e_src[n,h], e_dst[n,h] -------------------
// one wave per (node, head); C-length dot reduced across the wave
__global__ void att_scores(const float* __restrict__ h,
                           const float* __restrict__ att_src,
                           const float* __restrict__ att_dst,
                           float* __restrict__ esrc, float* __restrict__ edst,
                           int Nn, int H, int C) {
    const int wid = (int)(((size_t)blockIdx.x * blockDim.x + threadIdx.x) >> 5);
    const int lane = threadIdx.x & 31;
    if (wid >= Nn * H) return;
    const int n = wid / H, hh = wid % H;
    const float* base = h + (size_t)n * H * C + (size_t)hh * C;
    const float* as = att_src + hh * C;
    const float* ad = att_dst + hh * C;
    float ss = 0.0f, sd = 0.0f;
    for (int c = lane; c < C; c += 32) {
        const float v = base[c];
        ss += v * as[c];
        sd += v * ad[c];
    }
    for (int off = 16; off > 0; off >>= 1) {
        ss += __shfl_down(ss, off);
        sd += __shfl_down(sd, off);
    }
    if (lane == 0) { esrc[wid] = ss; edst[wid] = sd; }
}

// ---------------- edge softmax pass 1: per-(dst,head) max --------------------
__global__ void edge_max(const int* __restrict__ ei, int E, int Nn, int H,
                         const float* __restrict__ esrc,
                         const float* __restrict__ edst,
                         float* __restrict__ m) {
    const int e = blockIdx.x * blockDim.x + threadIdx.x;
    if (e >= E + Nn) return;
    int s, d;
    if (e < E) { s = ei[e]; d = ei[E + e]; } else { s = d = e - E; }
    for (int h = 0; h < H; ++h) {
        float v = esrc[s * H + h] + edst[d * H + h];
        v = (v > 0.0f) ? v : NEG_SLOPE * v;
        atomicMaxFloat(&m[d * H + h], v);
    }
}

// ---------------- edge pass 2: fused p, z-sum, unnormalized aggregate --------
// one block per edge; thread k handles channel k (k in [0, H*C))
__global__ void edge_aggregate(const int* __restrict__ ei, int E, int Nn,
                               int H, int C,
                               const float* __restrict__ esrc,
                               const float* __restrict__ edst,
                               const float* __restrict__ m,
                               const float* __restrict__ hfeat,
                               float* __restrict__ z, float* __restrict__ agg) {
    const int e = blockIdx.x;
    const int k = threadIdx.x;
    int s, d;
    if (e < E) { s = ei[e]; d = ei[E + e]; } else { s = d = e - E; }
    const int h = k / C;
    float v = esrc[s * H + h] + edst[d * H + h];
    v = (v > 0.0f) ? v : NEG_SLOPE * v;
    const float p = expf(v - m[d * H + h]);
    if ((k % C) == 0) atomicAdd(&z[d * H + h], p);
    atomicAdd(&agg[(size_t)d * H * C + k], p * hfeat[(size_t)s * H * C + k]);
}

// ---------------- per-node normalize by z, add GAT bias ----------------------
__global__ void norm_bias(float* __restrict__ agg, const float* __restrict__ z,
                          const float* __restrict__ bias, int Nn, int H, int C) {
    const size_t i = (size_t)blockIdx.x * blockDim.x + threadIdx.x;
    const int F = H * C;
    if (i >= (size_t)Nn * F) return;
    const int n = (int)(i / F);
    const int k = (int)(i % F);
    const int h = k / C;
    agg[i] = agg[i] / z[n * H + h] + bias[k];
}

// ---------------- BN stats: per-channel sum & sumsq (blockDim.x == F) --------
__global__ void bn_stats(const float* __restrict__ x, float* __restrict__ stats,
                         int Nn, int F) {
    const int c = threadIdx.x;
    const int chunk = (Nn + gridDim.x - 1) / gridDim.x;
    const int n0 = blockIdx.x * chunk;
    const int n1 = min(n0 + chunk, Nn);
    float s = 0.0f, ss = 0.0f;
    for (int n = n0; n < n1; ++n) {
        const float v = x[(size_t)n * F + c];
        s += v;
        ss += v * v;
    }
    atomicAdd(&stats[c], s);
    atomicAdd(&stats[F + c], ss);
}

// ---------------- BN apply + ELU (in place) ----------------------------------
__global__ void bn_elu(float* __restrict__ x, const float* __restrict__ stats,
                       const float* __restrict__ g, const float* __restrict__ b,
                       int Nn, int F) {
    const size_t i = (size_t)blockIdx.x * blockDim.x + threadIdx.x;
    if (i >= (size_t)Nn * F) return;
    const int c = (int)(i % F);
    const float mean = stats[c] / (float)Nn;
    const float var = stats[F + c] / (float)Nn - mean * mean;
    float v = g[c] * (x[i] - mean) * rsqrtf(var + BN_EPS) + b[c];
    x[i] = (v > 0.0f) ? v : (expf(v) - 1.0f);
}

// ---------------- final 32->2 linear + log_softmax ---------------------------
__global__ void head_kernel(const float* __restrict__ c1,
                            const float* __restrict__ Wc2,
                            const float* __restrict__ bc2,
                            float* __restrict__ out, int Nn) {
    const int n = blockIdx.x * blockDim.x + threadIdx.x;
    if (n >= Nn) return;
    float l0 = bc2[0], l1 = bc2[1];
    const float* row = c1 + (size_t)n * 32;
    for (int k = 0; k < 32; ++k) {
        const float v = row[k];
        l0 += v * Wc2[k * 2 + 0];
        l1 += v * Wc2[k * 2 + 1];
    }
    const float mx = fmaxf(l0, l1);
    const float lse = mx + logf(expf(l0 - mx) + expf(l1 - mx));
    out[n * 2 + 0] = l0 - lse;
    out[n * 2 + 1] = l1 - lse;
}

__global__ void fill_f32(float* __restrict__ p, float v, size_t n) {
    const size_t i = (size_t)blockIdx.x * blockDim.x + threadIdx.x;
    if (i < n) p[i] = v;
}

extern "C" void kernel_launch(void* const* d_in, const int* in_sizes, int n_in,
                              void* d_out, int out_size, void* d_ws, size_t ws_size,
                              hipStream_t stream) {
    const float* x   = (const float*)d_in[0];
    const int*   ei  = (const int*)d_in[1];
    const float* W1  = (const float*)d_in[2];
    const float* as1 = (const float*)d_in[3];
    const float* ad1 = (const float*)d_in[4];
    const float* b1  = (const float*)d_in[5];
    const float* g1  = (const float*)d_in[6];
    const float* be1 = (const float*)d_in[7];
    const float* W2  = (const float*)d_in[8];
    const float* as2 = (const float*)d_in[9];
    const float* ad2 = (const float*)d_in[10];
    const float* b2  = (const float*)d_in[11];
    const float* g2  = (const float*)d_in[12];
    const float* be2 = (const float*)d_in[13];
    const float* Wc1 = (const float*)d_in[14];
    const float* bc1 = (const float*)d_in[15];
    const float* Wc2 = (const float*)d_in[16];
    const float* bc2 = (const float*)d_in[17];
    float* out = (float*)d_out;

    // workspace carve-up
    float* ws   = (float*)d_ws;
    float* h1   = ws;                          // [N, 256]
    float* agg1 = h1   + (size_t)NODES * F1;   // [N, 256]
    float* h2   = agg1 + (size_t)NODES * F1;   // [N, 64]
    float* agg2 = h2   + (size_t)NODES * HIDC; // [N, 64]
    float* es1  = agg2 + (size_t)NODES * HIDC; // [N, 4]
    float* ed1  = es1 + (size_t)NODES * H1;
    float* m1   = ed1 + (size_t)NODES * H1;
    float* z1   = m1  + (size_t)NODES * H1;
    float* es2  = z1  + (size_t)NODES * H1;    // [N]
    float* ed2  = es2 + NODES;
    float* m2   = ed2 + NODES;
    float* z2   = m2  + NODES;
    float* stats = z2 + NODES;                 // 2*256 (reused as 2*64)
    float* c1 = h1;                            // reuse h1 for classifier hidden [N,32]

    const int MT = NODES / 16;                 // 6250 M-tiles, exact

    // ---- layer 1: h1 = x @ W1 (WMMA, LDS-staged B) ----
    wmma_gemm_f32<0><<<dim3(cdiv(MT, 8), F1 / 16), 256, 0, stream>>>(
        x, W1, nullptr, h1, DIN, F1, MT);

    // attention scores
    att_scores<<<cdiv((long long)NODES * H1 * 32, 256), 256, 0, stream>>>(
        h1, as1, ad1, es1, ed1, NODES, H1, HIDC);

    // init m/z/agg
    fill_f32<<<cdiv((long long)NODES * H1, 256), 256, 0, stream>>>(m1, -INFINITY, (size_t)NODES * H1);
    fill_f32<<<cdiv((long long)NODES * H1, 256), 256, 0, stream>>>(z1, 0.0f, (size_t)NODES * H1);
    fill_f32<<<cdiv((long long)NODES * F1, 256), 256, 0, stream>>>(agg1, 0.0f, (size_t)NODES * F1);

    edge_max<<<cdiv(ETOT, 256), 256, 0, stream>>>(ei, EDGES, NODES, H1, es1, ed1, m1);
    edge_aggregate<<<ETOT, F1, 0, stream>>>(ei, EDGES, NODES, H1, HIDC, es1, ed1, m1, h1, z1, agg1);
    norm_bias<<<cdiv((long long)NODES * F1, 256), 256, 0, stream>>>(agg1, z1, b1, NODES, H1, HIDC);

    // BN1 + ELU
    fill_f32<<<1, 2 * F1, 0, stream>>>(stats, 0.0f, (size_t)(2 * F1));
    bn_stats<<<256, F1, 0, stream>>>(agg1, stats, NODES, F1);
    bn_elu<<<cdiv((long long)NODES * F1, 256), 256, 0, stream>>>(agg1, stats, g1, be1, NODES, F1);

    // ---- layer 2: h2 = agg1 @ W2 (WMMA, LDS-staged B) ----
    wmma_gemm_f32<0><<<dim3(cdiv(MT, 8), HIDC / 16), 256, 0, stream>>>(
        agg1, W2, nullptr, h2, F1, HIDC, MT);

    att_scores<<<cdiv((long long)NODES * 32, 256), 256, 0, stream>>>(
        h2, as2, ad2, es2, ed2, NODES, 1, HIDC);

    fill_f32<<<cdiv(NODES, 256), 256, 0, stream>>>(m2, -INFINITY, (size_t)NODES);
    fill_f32<<<cdiv(NODES, 256), 256, 0, stream>>>(z2, 0.0f, (size_t)NODES);
    fill_f32<<<cdiv((long long)NODES * HIDC, 256), 256, 0, stream>>>(agg2, 0.0f, (size_t)NODES * HIDC);

    edge_max<<<cdiv(ETOT, 256), 256, 0, stream>>>(ei, EDGES, NODES, 1, es2, ed2, m2);
    edge_aggregate<<<ETOT, HIDC, 0, stream>>>(ei, EDGES, NODES, 1, HIDC, es2, ed2, m2, h2, z2, agg2);
    norm_bias<<<cdiv((long long)NODES * HIDC, 256), 256, 0, stream>>>(agg2, z2, b2, NODES, 1, HIDC);

    // BN2 + ELU
    fill_f32<<<1, 2 * HIDC, 0, stream>>>(stats, 0.0f, (size_t)(2 * HIDC));
    bn_stats<<<256, HIDC, 0, stream>>>(agg2, stats, NODES, HIDC);
    bn_elu<<<cdiv((long long)NODES * HIDC, 256), 256, 0, stream>>>(agg2, stats, g2, be2, NODES, HIDC);

    // ---- classifier: c1 = ELU(agg2 @ Wc1 + bc1) (WMMA, fused bias+ELU) ----
    wmma_gemm_f32<1><<<dim3(cdiv(MT, 8), 32 / 16), 256, 0, stream>>>(
        agg2, Wc1, bc1, c1, HIDC, 32, MT);

    // final linear + log_softmax
    head_kernel<<<cdiv(NODES, 256), 256, 0, stream>>>(c1, Wc2, bc2, out, NODES);
}